// PostProcessingModule_50577534878281
// MI455X (gfx1250) — compile-verified
//
#include <hip/hip_runtime.h>
#include <stdint.h>

// Problem constants (match reference)
#define B_   8
#define ND_  8
#define H_   512
#define W_   512
#define HW_  (H_ * W_)
#define K_   32
#define CONF_        0.9f
#define MARGIN2_     9.0f      // MARGIN^2
#define MIN_CLUSTER_ 50.0f

// Streaming parameters
#define SEG_CHUNK 2048                 // pixels per async seg chunk (8 KB)
#define NCHUNK    (HW_ / SEG_CHUNK)    // 128
#define PFD       512                  // prefetch distance in pixels

typedef float v2f __attribute__((ext_vector_type(2)));
typedef float v8f __attribute__((ext_vector_type(8)));

#define S_WAIT_ASYNCCNT(n) asm volatile("s_wait_asynccnt " #n ::: "memory")

__device__ __forceinline__ float readlane_f(float v, int lane) {
  return __int_as_float(__builtin_amdgcn_readlane(__float_as_int(v), lane));
}

// VALU-only 32-lane unsigned-min all-reduce:
// xor1/xor2 via DPP quad_perm, xor4/xor8 via DPP ROW_XMASK, xor16 via v_permlanex16.
__device__ __forceinline__ unsigned wave32_umin(unsigned pk) {
  unsigned t;
  t = (unsigned)__builtin_amdgcn_update_dpp(0, (int)pk, 0xB1,  0xf, 0xf, true); // quad_perm [1,0,3,2]
  pk = (t < pk) ? t : pk;
  t = (unsigned)__builtin_amdgcn_update_dpp(0, (int)pk, 0x4E,  0xf, 0xf, true); // quad_perm [2,3,0,1]
  pk = (t < pk) ? t : pk;
  t = (unsigned)__builtin_amdgcn_update_dpp(0, (int)pk, 0x164, 0xf, 0xf, true); // ROW_XMASK 4
  pk = (t < pk) ? t : pk;
  t = (unsigned)__builtin_amdgcn_update_dpp(0, (int)pk, 0x168, 0xf, 0xf, true); // ROW_XMASK 8
  pk = (t < pk) ? t : pk;
  t = (unsigned)__builtin_amdgcn_permlanex16(0, (int)pk, 0x76543210, 0xFEDCBA98,
                                             false, false);                     // half swap (xor16)
  pk = (t < pk) ? t : pk;
  return pk;
}

// Async-copy one SEG_CHUNK of floats (gsrc -> LDS) using 16x global_load_async_to_lds_b128.
// Each issue moves 32 lanes * 16B = 128 pixels. ASYNCcnt += 16.
__device__ __forceinline__ void async_seg_chunk(const float* gsrc, float* lds_dst, int lane) {
#pragma unroll
  for (int i = 0; i < SEG_CHUNK; i += 128) {
    uint32_t laddr = (uint32_t)(uintptr_t)(lds_dst + i + lane * 4);
    const float* g = gsrc + i + lane * 4;
    asm volatile("global_load_async_to_lds_b128 %0, %1, off"
                 :: "v"(laddr), "v"(g)
                 : "memory");
  }
}

__device__ __forceinline__ v8f wmma_f32_4(float a0, float a1, float bb0, float bb1, v8f c) {
  v2f A;  A[0] = a0;  A[1] = a1;
  v2f Bm; Bm[0] = bb0; Bm[1] = bb1;
  // 8 args: (neg_a, A, neg_b, B, c_mod, C, reuse_a, reuse_b)
  return __builtin_amdgcn_wmma_f32_16x16x4_f32(false, A, false, Bm, (short)0, c, false, false);
}

// ---------------------------------------------------------------------------
// Kernel 0: zero the accumulators
// ---------------------------------------------------------------------------
__global__ void zero_kernel(float* __restrict__ p, int n) {
  int i = blockIdx.x * blockDim.x + threadIdx.x;
  if (i < n) p[i] = 0.f;
}

// ---------------------------------------------------------------------------
// Kernel 1: sequential greedy clustering scan, one wave32 per batch.
// lane == cluster id. Centers live in WMMA B-matrix layout:
//   b1_* : clusters 0..15  (b1_0/b1_1 = dims k0,k1 in lanes n<16; k2,k3 in lanes n+16;
//                           b1_2/b1_3 = dims k4,k5 / k6,k7 likewise)
//   b2_* : clusters 16..31 (same layout, N = cluster-16)
// Per valid pixel: 4x v_wmma_f32_16x16x4_f32 -> dot(c_lane, e) for all 32 lanes.
// Segment sums accumulated directly via global atomics (lane 0).
// ---------------------------------------------------------------------------
__global__ __launch_bounds__(32, 1)
void cluster_scan_kernel(const float* __restrict__ seg,
                         const float* __restrict__ emb,
                         const float* __restrict__ offp,
                         const float* __restrict__ zp,
                         float* __restrict__ acc) {
  const int b    = blockIdx.x;
  const int lane = threadIdx.x;

  const float* segb = seg  + (size_t)b * HW_;
  const float* embb = emb  + (size_t)b * ND_ * HW_;
  const float* offb = offp + (size_t)b * HW_;
  const float* zb   = zp   + (size_t)b * HW_;

  float* accC = acc + (size_t)(b * K_) * H_;
  float* accX = acc + (size_t)B_ * K_ * H_     + (size_t)(b * K_) * H_;
  float* accZ = acc + (size_t)2 * B_ * K_ * H_ + (size_t)(b * K_) * H_;

  __shared__ __align__(16) float seg_lds[2][SEG_CHUNK];

  // Prime the double buffer.
  async_seg_chunk(segb,             seg_lds[0], lane);
  async_seg_chunk(segb + SEG_CHUNK, seg_lds[1], lane);

  float b1_0 = 0.f, b1_1 = 0.f, b1_2 = 0.f, b1_3 = 0.f;
  float b2_0 = 0.f, b2_1 = 0.f, b2_2 = 0.f, b2_3 = 0.f;
  float cnt = 0.f;   // lane == cluster: member count
  float cn2 = 0.f;   // lane == cluster: ||center||^2
  int   nact = 0;    // uniform: number of active clusters (activated in order 0,1,2,...)
  const bool hi     = (lane >= 16);
  const int  lane15 = lane & 15;

  for (int c = 0; c < NCHUNK; ++c) {
    if (c + 1 < NCHUNK) { S_WAIT_ASYNCCNT(16); } else { S_WAIT_ASYNCCNT(0); }
    const float* sl    = seg_lds[c & 1];
    const int pchunk   = c * SEG_CHUNK;

    for (int g0 = 0; g0 < SEG_CHUNK; g0 += 32) {
      const int p0 = pchunk + g0;

      // Warm caches PFD pixels ahead: lanes 0..7 -> 8 embedding rows,
      // lane 8 -> offset line, lane 9 -> z line. One prefetch instruction.
      {
        int pf = p0 + PFD;
        if (pf < HW_) {
          const float* pa = (lane < 8)  ? (embb + (size_t)lane * HW_ + pf)
                          : (lane == 8) ? (offb + pf)
                          : (lane == 9) ? (zb + pf)
                                        : (segb + pf);
          __builtin_prefetch(pa, 0, 1);
        }
      }

      float s = sl[g0 + lane];
      unsigned mask = __builtin_amdgcn_ballot_w32(s >= CONF_);

      while (mask) {
        const int j = __builtin_ctz(mask);
        mask &= mask - 1;
        const int p = p0 + j;

        // Point data (uniform address / per-lane dim); prefetched above.
        float ev   = embb[(size_t)(lane & 7) * HW_ + p];
        float offv = offb[p];
        float zv   = zb[p];

        float e0 = readlane_f(ev, 0), e1 = readlane_f(ev, 1);
        float e2 = readlane_f(ev, 2), e3 = readlane_f(ev, 3);
        float e4 = readlane_f(ev, 4), e5 = readlane_f(ev, 5);
        float e6 = readlane_f(ev, 6), e7 = readlane_f(ev, 7);

        // A-matrix (16x4, all rows = e[k..k+3]): per documented layout,
        // lanes<16 hold K=0,1 in v0,v1; lanes>=16 hold K=2,3.
        float a0 = hi ? e2 : e0;
        float a1 = hi ? e3 : e1;
        float a2 = hi ? e6 : e4;
        float a3 = hi ? e7 : e5;

        v8f D1 = {0.f,0.f,0.f,0.f,0.f,0.f,0.f,0.f};
        v8f D2 = {0.f,0.f,0.f,0.f,0.f,0.f,0.f,0.f};
        D1 = wmma_f32_4(a0, a1, b1_0, b1_1, D1);   // clusters 0..15, k0..3
        D1 = wmma_f32_4(a2, a3, b1_2, b1_3, D1);   // + k4..7
        D2 = wmma_f32_4(a0, a1, b2_0, b2_1, D2);   // clusters 16..31
        D2 = wmma_f32_4(a2, a3, b2_2, b2_3, D2);

        // D layout: v0 lane l<16 -> D[M=0][N=l]; lane l>=16 -> D[M=8][N=l-16].
        // All rows identical => dot for cluster == lane:
        float dotv = hi ? D2[0] : D1[0];

        float e2sum = e0*e0 + e1*e1 + e2*e2 + e3*e3 + e4*e4 + e5*e5 + e6*e6 + e7*e7;
        float dist  = fmaxf(cn2 - 2.f * dotv + e2sum, 0.f);
        // Active clusters are exactly lanes < nact (activation is in order).
        float distm = (lane < nact) ? dist : __builtin_inff();

        // argmin over 32 lanes: monotonic uint trick, low 5 bits = lane.
        unsigned pk = (__float_as_uint(distm) & 0xFFFFFFE0u) | (unsigned)lane;
        pk = wave32_umin(pk);

        unsigned pku = (unsigned)__builtin_amdgcn_readfirstlane((int)pk);
        int   cid_min = (int)(pku & 31u);
        float dmin    = readlane_f(distm, cid_min);
        bool  join    = dmin < MARGIN2_;

        int cap = (nact < K_ - 1) ? nact : (K_ - 1);
        int cid = join ? cid_min : cap;
        cid = __builtin_amdgcn_readfirstlane(cid);
        if (!join && nact < K_) nact++;   // uniform update

        float cnt_c = readlane_f(cnt, cid);
        float rcp   = __builtin_amdgcn_rcpf(cnt_c + 1.f);  // exact small-int denominator
        bool  updl  = (lane15 == (cid & 15));  // the 2 lanes holding this cluster's dims

        if (cid < 16) {  // uniform branch
          float n0 = join ? (b1_0 * cnt_c + a0) * rcp : a0;
          float n1 = join ? (b1_1 * cnt_c + a1) * rcp : a1;
          float n2 = join ? (b1_2 * cnt_c + a2) * rcp : a2;
          float n3 = join ? (b1_3 * cnt_c + a3) * rcp : a3;
          b1_0 = updl ? n0 : b1_0;
          b1_1 = updl ? n1 : b1_1;
          b1_2 = updl ? n2 : b1_2;
          b1_3 = updl ? n3 : b1_3;
        } else {
          float n0 = join ? (b2_0 * cnt_c + a0) * rcp : a0;
          float n1 = join ? (b2_1 * cnt_c + a1) * rcp : a1;
          float n2 = join ? (b2_2 * cnt_c + a2) * rcp : a2;
          float n3 = join ? (b2_3 * cnt_c + a3) * rcp : a3;
          b2_0 = updl ? n0 : b2_0;
          b2_1 = updl ? n1 : b2_1;
          b2_2 = updl ? n2 : b2_2;
          b2_3 = updl ? n3 : b2_3;
        }

        // Exact new ||center||^2 from the stored dims (lanes cid&15 and +16).
        float s0 = (cid < 16) ? b1_0 : b2_0;
        float s1 = (cid < 16) ? b1_1 : b2_1;
        float s2 = (cid < 16) ? b1_2 : b2_2;
        float s3 = (cid < 16) ? b1_3 : b2_3;
        float pcs  = s0*s0 + s1*s1 + s2*s2 + s3*s3;
        float ncn2 = readlane_f(pcs, cid & 15) + readlane_f(pcs, (cid & 15) + 16);

        bool isc = (lane == cid);
        cn2 = isc ? ncn2 : cn2;
        cnt = isc ? (cnt_c + 1.f) : cnt;

        // Segment-sum accumulation (only valid pixels contribute to k>=1 segments).
        if (lane == 0) {
          int h = p >> 9;           // / W_
          int x = p & (W_ - 1);
          float sig  = __builtin_amdgcn_rcpf(1.f + __expf(-offv));
          float xadj = (float)x + sig;
          atomicAdd(accC + cid * H_ + h, 1.f);
          atomicAdd(accX + cid * H_ + h, xadj);
          atomicAdd(accZ + cid * H_ + h, zv);
        }
      } // valid-pixel loop
    }   // groups in chunk

    if (c + 2 < NCHUNK)
      async_seg_chunk(segb + (size_t)(c + 2) * SEG_CHUNK, seg_lds[c & 1], lane);
  }     // chunks
}

// ---------------------------------------------------------------------------
// Kernel 2: finalize. One block per (b, k); thread = h.
// ---------------------------------------------------------------------------
__global__ __launch_bounds__(H_)
void finalize_kernel(const float* __restrict__ acc, float* __restrict__ out) {
  const int bk = blockIdx.x;
  const int b  = bk / K_;
  const int k  = bk % K_;
  const int h  = threadIdx.x;

  const float* accC = acc;
  const float* accX = acc + (size_t)B_ * K_ * H_;
  const float* accZ = acc + (size_t)2 * B_ * K_ * H_;

  float c  = accC[(size_t)bk * H_ + h];
  float sx = accX[(size_t)bk * H_ + h];
  float sz = accZ[(size_t)bk * H_ + h];

  __shared__ float r1[H_];
  __shared__ float r2[H_];
  r1[h] = c;
  r2[h] = (c > 0.f) ? 1.f : 0.f;
  __syncthreads();
#pragma unroll
  for (int s = H_ / 2; s > 0; s >>= 1) {
    if (h < s) { r1[h] += r1[h + s]; r2[h] += r2[h + s]; }
    __syncthreads();
  }
  float size  = r1[0];
  float nrows = r2[0];

  bool valid = (c > 0.f) && (size >= MIN_CLUSTER_) && (nrows >= 2.f);
  float cd = fmaxf(c, 1.f);
  float mx = sx / cd;
  float mz = sz / cd;
  float xw = (512.f - ((float)h + 0.5f)) * 0.2f;   // (MAX_X/MPP0 - (h+0.5))*MPP0
  float yw = -(mx - (float)(W_ / 2)) * 0.2f;

  // points: (B, 3, K, H) then valid: (B, K, H), concatenated flat.
  out[(((size_t)b * 3 + 0) * K_ + k) * H_ + h] = valid ? xw : 0.f;
  out[(((size_t)b * 3 + 1) * K_ + k) * H_ + h] = valid ? yw : 0.f;
  out[(((size_t)b * 3 + 2) * K_ + k) * H_ + h] = valid ? mz : 0.f;
  out[(size_t)B_ * 3 * K_ * H_ + (size_t)bk * H_ + h] = valid ? 1.f : 0.f;
}

// ---------------------------------------------------------------------------
extern "C" void kernel_launch(void* const* d_in, const int* in_sizes, int n_in,
                              void* d_out, int out_size, void* d_ws, size_t ws_size,
                              hipStream_t stream) {
  (void)in_sizes; (void)n_in; (void)out_size; (void)ws_size;
  const float* seg  = (const float*)d_in[0];
  const float* emb  = (const float*)d_in[1];
  const float* offp = (const float*)d_in[2];
  const float* zp   = (const float*)d_in[3];
  // d_in[4] (intrinsic) and d_in[5] (extrinsic) are unused by the reference.

  float* acc = (float*)d_ws;               // [3][B][K][H] accumulators
  const int accN = 3 * B_ * K_ * H_;

  zero_kernel<<<(accN + 255) / 256, 256, 0, stream>>>(acc, accN);
  cluster_scan_kernel<<<B_, 32, 0, stream>>>(seg, emb, offp, zp, acc);
  finalize_kernel<<<B_ * K_, H_, 0, stream>>>(acc, (float*)d_out);
}